// HYBLayer_88072599371931
// MI455X (gfx1250) — compile-verified
//
#include <hip/hip_runtime.h>
#include <hip/hip_bf16.h>

#define N_NODES 50000
#define N_EDGES 1600000
#define IN_DIM 128
#define HID_DIM 64
#define N_CH 6

typedef __attribute__((ext_vector_type(2))) float v2f;
typedef __attribute__((ext_vector_type(8))) float v8f;

// ---------------------------------------------------------------------------
// Dense projection h = x @ W  via V_WMMA_F32_16X16X4_F32 (full f32 precision).
// One workgroup = 4 waves; wave w computes the 16x16 tile at (m0, 16*w).
// A layout (16x4 f32): lanes 0-15 -> M=lane, K=k..k+1 ; lanes 16-31 -> K=k+2..k+3
// B layout (4x16 f32): lanes 0-15 -> N=lane, K=k..k+1 ; lanes 16-31 -> K=k+2..k+3
// C/D layout: VGPR v -> M = v (lanes 0-15) / v+8 (lanes 16-31), N = lane&15
// ---------------------------------------------------------------------------
__global__ __launch_bounds__(128) void proj_wmma_f32(
    const float* __restrict__ x,     // [N_NODES, IN_DIM]
    const float* __restrict__ Wc,    // [IN_DIM, HID_DIM] (one channel)
    float* __restrict__ h)           // [N_NODES, HID_DIM]
{
    const int wave = threadIdx.x >> 5;
    const int lane = threadIdx.x & 31;
    const int m0   = blockIdx.x * 16;      // 3125 blocks exactly cover 50000 rows
    const int n0   = wave * 16;            // 4 waves cover HID_DIM=64
    const int half = lane >> 4;            // 0: K+{0,1}, 1: K+{2,3}
    const int l15  = lane & 15;

    const float* xrow = x  + (size_t)(m0 + l15) * IN_DIM + half * 2;
    const float* wcol = Wc + (size_t)(half * 2) * HID_DIM + (n0 + l15);

    v8f c = {};
#pragma unroll 8
    for (int k = 0; k < IN_DIM; k += 4) {
        v2f a, b;
        a.x = xrow[k];
        a.y = xrow[k + 1];
        b.x = wcol[(size_t)k * HID_DIM];
        b.y = wcol[(size_t)(k + 1) * HID_DIM];
        c = __builtin_amdgcn_wmma_f32_16x16x4_f32(
                /*neg_a=*/false, a, /*neg_b=*/false, b,
                /*c_mod=*/(short)0, c, /*reuse_a=*/false, /*reuse_b=*/false);
    }

    float* hp = h + (size_t)(m0 + half * 8) * HID_DIM + (n0 + l15);
#pragma unroll
    for (int v = 0; v < 8; v++)
        hp[(size_t)v * HID_DIM] = c[v];
}

// ---------------------------------------------------------------------------
// Edge-parallel SpMM: hout[dst] += w * hin[src].  One wave per edge,
// float2 per lane (64 feats), f32 HW atomics (L2-resident scatter).
// ---------------------------------------------------------------------------
__global__ __launch_bounds__(256) void spmm_edges(
    const float* __restrict__ hin, float* __restrict__ hout,
    const int* __restrict__ src, const int* __restrict__ dst,
    const float* __restrict__ w, int n_edges)
{
    const int edge = blockIdx.x * 8 + (threadIdx.x >> 5);
    if (edge >= n_edges) return;
    const int lane = threadIdx.x & 31;

    const int   s  = src[edge];
    const int   d  = dst[edge];
    const float wt = w[edge];

    const float2 v = ((const float2*)(hin + (size_t)s * HID_DIM))[lane];
    float* ho = hout + (size_t)d * HID_DIM + lane * 2;
    unsafeAtomicAdd(ho + 0, wt * v.x);
    unsafeAtomicAdd(ho + 1, wt * v.y);
}

__global__ void fill_zero4(float4* __restrict__ p, int n4)
{
    int i = blockIdx.x * blockDim.x + threadIdx.x;
    if (i < n4) p[i] = make_float4(0.f, 0.f, 0.f, 0.f);
}

// out[:, ch*64:(ch+1)*64] = relu(h1 - (h2?) + bias)
__global__ void finalize_channel(
    const float* __restrict__ h1, const float* __restrict__ h2,
    const float* __restrict__ bias, float* __restrict__ out, int ch)
{
    int idx = blockIdx.x * blockDim.x + threadIdx.x;
    if (idx >= N_NODES * HID_DIM) return;
    const int node = idx >> 6;
    const int f    = idx & 63;
    float v = h1[idx];
    if (h2) v -= h2[idx];
    v += bias[f];
    out[(size_t)node * (N_CH * HID_DIM) + ch * HID_DIM + f] = fmaxf(v, 0.f);
}

// ---------------------------------------------------------------------------
// Orchestration
// ---------------------------------------------------------------------------
extern "C" void kernel_launch(void* const* d_in, const int* in_sizes, int n_in,
                              void* d_out, int out_size, void* d_ws, size_t ws_size,
                              hipStream_t stream)
{
    (void)in_sizes; (void)n_in; (void)out_size; (void)ws_size;

    const float* x     = (const float*)d_in[0];   // [N, 128]
    const int*   esrc  = (const int*)  d_in[1];   // [E]
    const int*   edst  = (const int*)  d_in[2];   // [E]
    const float* gcn_w = (const float*)d_in[3];   // [E]
    const float* sct_w = (const float*)d_in[4];   // [E]
    const float* W     = (const float*)d_in[5];   // [6, 128, 64]
    const float* bias  = (const float*)d_in[6];   // [6, 1, 64]
    float*       out   = (float*)d_out;           // [N, 384]

    const size_t FEAT = (size_t)N_NODES * HID_DIM;   // 3.2M floats = 12.8 MB
    float* buf[3] = { (float*)d_ws,
                      (float*)d_ws + FEAT,
                      (float*)d_ws + 2 * FEAT };

    const dim3 projGrid(N_NODES / 16, 1, 1);
    const int  spmmBlocks = (N_EDGES + 7) / 8;
    const int  zeroBlocks = (int)((FEAT / 4 + 255) / 256);
    const int  finBlocks  = (int)((FEAT + 255) / 256);

    const int CONFIG[N_CH] = { -1, -2, -3, 1, 2, 3 };

    auto zero = [&](float* p) {
        fill_zero4<<<zeroBlocks, 256, 0, stream>>>((float4*)p, (int)(FEAT / 4));
    };
    auto spmm = [&](const float* hin, float* hout, const float* w) {
        zero(hout);
        spmm_edges<<<spmmBlocks, 256, 0, stream>>>(hin, hout, esrc, edst, w, N_EDGES);
    };

    for (int ch = 0; ch < N_CH; ch++) {
        const int scale = CONFIG[ch];

        // h = x @ W[ch]  -> buf[0]
        proj_wmma_f32<<<projGrid, 128, 0, stream>>>(
            x, W + (size_t)ch * IN_DIM * HID_DIM, buf[0]);

        if (scale < 0) {
            // GCN channel: apply T |scale| times
            float* cur = buf[0];
            float* nxt = buf[1];
            for (int t = 0; t < -scale; t++) {
                spmm(cur, nxt, gcn_w);
                float* tmp = cur; cur = nxt; nxt = tmp;
            }
            finalize_channel<<<finBlocks, 256, 0, stream>>>(
                cur, nullptr, bias + (size_t)ch * HID_DIM, out, ch);
        } else {
            // Scattering channel: h1 = T^p h ; h2 = T^p h1 ; out = h1 - h2
            const int p = 1 << (scale - 1);
            float* cur = buf[0];
            float* nxt = buf[1];
            for (int t = 0; t < p; t++) {
                spmm(cur, nxt, sct_w);
                float* tmp = cur; cur = nxt; nxt = tmp;
            }
            float* h1 = cur;                       // preserved below
            float* s0 = (h1 == buf[0]) ? buf[1] : buf[0];
            float* s1 = buf[2];
            float* c2 = h1;
            float* n2 = s0;
            for (int t = 0; t < p; t++) {
                spmm(c2, n2, sct_w);
                c2 = n2;
                n2 = (c2 == s0) ? s1 : s0;
            }
            finalize_channel<<<finBlocks, 256, 0, stream>>>(
                h1, c2, bias + (size_t)ch * HID_DIM, out, ch);
        }
    }
}